// EnergyEwald_23613730193756
// MI455X (gfx1250) — compile-verified
//
#include <hip/hip_runtime.h>
#include <hip/hip_bf16.h>
#include <cmath>

typedef __attribute__((ext_vector_type(2))) float v2f;
typedef __attribute__((ext_vector_type(8))) float v8f;

#define N_ATOMS_C 100000
#define N_PAIRS_C 6400000
#define N_MOL_C   64
#define NKPAD     1024
#define KE_C      14.3996f
#define ALPHA_C   0.3f
#define CUTOFF_C  10.0f
#define TWO_PI_C  6.28318530717958647692f

// ---------------------------------------------------------------- init ------
__global__ void k_init(float* out, int* mstart, int* mend) {
    int t = threadIdx.x;
    if (t < N_MOL_C) { out[t] = 0.0f; mstart[t] = N_ATOMS_C; mend[t] = 0; }
}

// ------------------------------------------- atom pass: bounds + self-term --
__global__ void __launch_bounds__(256) k_atoms(const float* __restrict__ q,
                                               const int* __restrict__ idx_m,
                                               int* mstart, int* mend, float* out) {
    __shared__ float bin[N_MOL_C];
    if (threadIdx.x < N_MOL_C) bin[threadIdx.x] = 0.0f;
    __syncthreads();
    int stride = gridDim.x * blockDim.x;
    for (int a = blockIdx.x * blockDim.x + threadIdx.x; a < N_ATOMS_C; a += stride) {
        int m = idx_m[a];
        atomicMin(&mstart[m], a);
        atomicMax(&mend[m], a + 1);
        float qa = q[a];
        atomicAdd(&bin[m], qa * qa);
    }
    __syncthreads();
    if (threadIdx.x < N_MOL_C) {
        float v = bin[threadIdx.x];
        if (v != 0.0f)
            atomicAdd(&out[threadIdx.x],
                      -KE_C * sqrtf(ALPHA_C / 3.14159265358979323846f) * v);
    }
}

// ------------------------------- per-molecule kv (padded) + gfac -----------
__global__ void k_setup(const float* __restrict__ cell, const float* __restrict__ kvecs,
                        int nk, float* __restrict__ kvpad, float* __restrict__ gfac) {
    int m = blockIdx.x;
    const float* C = cell + (size_t)m * 9;
    float c00=C[0], c01=C[1], c02=C[2];
    float c10=C[3], c11=C[4], c12=C[5];
    float c20=C[6], c21=C[7], c22=C[8];
    float det = c00*(c11*c22 - c12*c21) - c01*(c10*c22 - c12*c20) + c02*(c10*c21 - c11*c20);
    float inv = 1.0f / det;
    // inverse rows (inv[e][d])
    float i00 =  (c11*c22 - c12*c21)*inv, i01 = -(c01*c22 - c02*c21)*inv, i02 =  (c01*c12 - c02*c11)*inv;
    float i10 = -(c10*c22 - c12*c20)*inv, i11 =  (c00*c22 - c02*c20)*inv, i12 = -(c00*c12 - c02*c10)*inv;
    float i20 =  (c10*c21 - c11*c20)*inv, i21 = -(c00*c21 - c01*c20)*inv, i22 =  (c00*c11 - c01*c10)*inv;
    float pref = KE_C * TWO_PI_C / fabsf(det);
    for (int k = threadIdx.x; k < NKPAD; k += blockDim.x) {
        float kx = 0.0f, ky = 0.0f, kz = 0.0f, g = 0.0f;
        if (k < nk) {
            float g0 = kvecs[3*k], g1 = kvecs[3*k+1], g2 = kvecs[3*k+2];
            // kv[e] = 2*pi * sum_d kvec[d] * inv[e][d]
            kx = TWO_PI_C * (g0*i00 + g1*i01 + g2*i02);
            ky = TWO_PI_C * (g0*i10 + g1*i11 + g2*i12);
            kz = TWO_PI_C * (g0*i20 + g1*i21 + g2*i22);
            float ksq = kx*kx + ky*ky + kz*kz;
            g = pref * __expf(-0.25f * ksq / ALPHA_C) / ksq;
        }
        float* p = kvpad + ((size_t)m * NKPAD + k) * 4;
        p[0] = kx; p[1] = ky; p[2] = kz; p[3] = 0.0f;
        gfac[(size_t)m * NKPAD + k] = g;
    }
}

// --------------------------------------------------- real-space pairs ------
// Rij / idx_i / idx_j are 128+ MB one-shot streams -> non-temporal loads so
// the 192MB L2 keeps q[] and the atom gathers resident; prefetch a stride
// ahead (global_prefetch_b8).
__global__ void __launch_bounds__(256) k_real(const float* __restrict__ q,
                                              const float* __restrict__ Rij,
                                              const int* __restrict__ idx_m,
                                              const int* __restrict__ idx_i,
                                              const int* __restrict__ idx_j,
                                              float frcut, float* out) {
    __shared__ float bin[N_MOL_C];
    if (threadIdx.x < N_MOL_C) bin[threadIdx.x] = 0.0f;
    __syncthreads();
    const float sqa = 0.54772255750516611f;   // sqrt(0.3)
    int stride = gridDim.x * blockDim.x;
    for (int p = blockIdx.x * blockDim.x + threadIdx.x; p < N_PAIRS_C; p += stride) {
        int pn = p + stride;
        if (pn < N_PAIRS_C) {
            __builtin_prefetch(&Rij[3 * (size_t)pn], 0, 0);
            __builtin_prefetch(&idx_i[pn], 0, 0);
            __builtin_prefetch(&idx_j[pn], 0, 0);
        }
        float rx = __builtin_nontemporal_load(&Rij[3*p]);
        float ry = __builtin_nontemporal_load(&Rij[3*p+1]);
        float rz = __builtin_nontemporal_load(&Rij[3*p+2]);
        float d2 = rx*rx + ry*ry + rz*rz;
        if (d2 <= CUTOFF_C * CUTOFF_C) {
            int i = __builtin_nontemporal_load(&idx_i[p]);
            int j = __builtin_nontemporal_load(&idx_j[p]);
            float d = sqrtf(d2);
            float f = erfcf(sqa * d) / d - frcut;
            atomicAdd(&bin[idx_m[i]], q[i] * q[j] * f);
        }
    }
    __syncthreads();
    if (threadIdx.x < N_MOL_C) {
        float v = bin[threadIdx.x];
        if (v != 0.0f) atomicAdd(&out[threadIdx.x], 0.5f * KE_C * v);
    }
}

// -------------------------- reciprocal space: WMMA f32 16x16x4 phase GEMM --
// grid (N_MOL, 8) x 256 threads; wave w owns k-tile (blockIdx.y*8 + w).
// Software-pipelined: tile t+1 loads issue before tile t's trans block.
// Pad columns keep garbage positions but qa=0, so they contribute nothing.
__global__ void __launch_bounds__(256) k_recip(const float* __restrict__ q,
                                               const float* __restrict__ R,
                                               const float* __restrict__ kvpad,
                                               const float* __restrict__ gfac,
                                               const int* __restrict__ mstart,
                                               const int* __restrict__ mend,
                                               float* out) {
    const int m    = blockIdx.x;
    const int wave = threadIdx.x >> 5;
    const int lane = threadIdx.x & 31;
    const int half = lane >> 4;        // 0: lanes 0-15, 1: lanes 16-31
    const int lidx = lane & 15;
    const int kbase = (blockIdx.y * 8 + wave) * 16;   // 16-k tile base, < 1024

    // A operand (16x4 f32): lane holds k-row M=lidx; VGPR0 = K{0|2}, VGPR1 = K{1|3}
    const float* kvr = kvpad + ((size_t)m * NKPAD + kbase + lidx) * 4;
    v2f A;
    A.x = half ? kvr[2] : kvr[0];
    A.y = half ? 0.0f   : kvr[1];

    const int s = mstart[m];
    const int e = mend[m];
    const int nt = (e > s) ? ((e - s + 15) >> 4) : 0;   // atom tiles

    float accre[8], accim[8];
#pragma unroll
    for (int v = 0; v < 8; ++v) { accre[v] = 0.0f; accim[v] = 0.0f; }
    v8f czero = {};

    if (nt > 0) {
        // prologue: unconditional clamped loads for tile 0
        int a  = s + lidx;
        int ac = (a < e) ? a : (e - 1);
        float rx = R[3*ac], ry = R[3*ac+1], rz = R[3*ac+2];
        float qa = q[ac];
        qa = (a < e) ? qa : 0.0f;

        for (int it = 0; it < nt; ++it) {
            // issue next tile's loads first (clamped; last iter's are discarded)
            int an  = s + (it + 1) * 16 + lidx;
            int anc = (an < e) ? an : (e - 1);
            float nrx = R[3*anc], nry = R[3*anc+1], nrz = R[3*anc+2];
            float nqa = q[anc];

            // B operand (4x16 f32): VGPR0 = row K{0|2} (x|z), VGPR1 = row K{1|3} (y|0)
            v2f B;
            B.x = half ? rz : rx;
            B.y = half ? 0.0f : ry;

            // kdp tile: rows = 16 k-vectors, cols = 16 atoms (f32, full precision)
            v8f c = __builtin_amdgcn_wmma_f32_16x16x4_f32(
                        false, A, false, B, (short)0, czero, false, false);

#pragma unroll
            for (int v = 0; v < 8; ++v) {
                float kd = c[v];       // k = kbase + v + 8*half, col = lidx
                accre[v] = __builtin_fmaf(qa, __cosf(kd), accre[v]);
                accim[v] = __builtin_fmaf(qa, __sinf(kd), accim[v]);
            }

            // rotate
            rx = nrx; ry = nry; rz = nrz;
            qa = (an < e) ? nqa : 0.0f;
        }
    }

    // reduce 16 atom-columns inside each half-wave (xor masks < 16 stay in-half)
#pragma unroll
    for (int v = 0; v < 8; ++v) {
#pragma unroll
        for (int off = 1; off < 16; off <<= 1) {
            accre[v] += __shfl_xor(accre[v], off, 32);
            accim[v] += __shfl_xor(accim[v], off, 32);
        }
    }

    float epart = 0.0f;
    if (lidx == 0) {                   // lanes 0 and 16 hold the 16 k sums
#pragma unroll
        for (int v = 0; v < 8; ++v) {
            int k = kbase + v + 8 * half;
            float g = gfac[(size_t)m * NKPAD + k];   // 0 for pad rows
            epart += g * (accre[v]*accre[v] + accim[v]*accim[v]);
        }
    }
    epart += __shfl_xor(epart, 16, 32);
    if (lane == 0 && epart != 0.0f) atomicAdd(&out[m], epart);
}

// ---------------------------------------------------------------------------
extern "C" void kernel_launch(void* const* d_in, const int* in_sizes, int n_in,
                              void* d_out, int out_size, void* d_ws, size_t ws_size,
                              hipStream_t stream) {
    const float* q     = (const float*)d_in[0];   // [A,1]
    const float* Rij   = (const float*)d_in[1];   // [P,3]
    const float* R     = (const float*)d_in[2];   // [A,3]
    const float* cell  = (const float*)d_in[3];   // [M,3,3]
    const float* kvecs = (const float*)d_in[4];   // [NK,3]
    const int*   idx_m = (const int*)d_in[5];     // [A]
    const int*   idx_i = (const int*)d_in[6];     // [P]
    const int*   idx_j = (const int*)d_in[7];     // [P]
    float* out = (float*)d_out;                   // [64]
    int nk = in_sizes[4] / 3;                     // 1020

    // workspace layout (~1.31 MB): kvpad [64][1024][4] f32, gfac [64][1024] f32,
    // mstart[64] i32, mend[64] i32
    float* kvpad  = (float*)d_ws;
    float* gfac   = kvpad + (size_t)N_MOL_C * NKPAD * 4;
    int*   mstart = (int*)(gfac + (size_t)N_MOL_C * NKPAD);
    int*   mend   = mstart + N_MOL_C;

    float frcut = erfcf(sqrtf(ALPHA_C) * CUTOFF_C) / CUTOFF_C;

    k_init <<<1, 64, 0, stream>>>(out, mstart, mend);
    k_atoms<<<256, 256, 0, stream>>>(q, idx_m, mstart, mend, out);
    k_setup<<<N_MOL_C, 256, 0, stream>>>(cell, kvecs, nk, kvpad, gfac);
    k_real <<<2048, 256, 0, stream>>>(q, Rij, idx_m, idx_i, idx_j, frcut, out);
    dim3 grecip(N_MOL_C, 8);
    k_recip<<<grecip, 256, 0, stream>>>(q, R, kvpad, gfac, mstart, mend, out);
}